// SimpleDiagonalRNN_84232898609663
// MI455X (gfx1250) — compile-verified
//
#include <hip/hip_runtime.h>
#include <hip/hip_bf16.h>
#include <stdint.h>

// SimpleDiagonalRNN: h_t = a*h_{t-1} + x_t, a = 1 - relu(w), x:[8,4096,512] f32.
// Memory-bound: 134MB min traffic @ 23.3 TB/s ~= 5.8us. 3-phase chunked scan:
//   A) per-(b,chunk,d) local terminal values (reads x once, warms 192MB L2)
//   B) tiny in-place scan of chunk carries with multiplier a^64
//   C) exact scan per chunk; x tile staged into LDS by the CDNA5 Tensor Data
//      Mover, double-buffered in two 32-row subtiles with partial
//      s_wait_tensorcnt so stores of subtile 0 overlap the DMA of subtile 1.

#define BB    8
#define TT    4096
#define DD    512
#define NC    64          // chunks along t
#define LL    64          // TT / NC
#define HLL   32          // subtile rows (double buffer)
#define DTILE 256         // d-columns per phase-C block

typedef __attribute__((ext_vector_type(4))) unsigned int v4u;
typedef __attribute__((ext_vector_type(4))) int          v4i;
typedef __attribute__((ext_vector_type(8))) int          v8i;

#if defined(__has_builtin)
#if __has_builtin(__builtin_amdgcn_tensor_load_to_lds)
#define HAVE_TDM 1
#endif
#endif

#ifdef HAVE_TDM
// Issue one TDM 2D tile load: rows x DTILE f32 elements, row stride = DD elems.
// Must be executed by a single wave; tracked with TENSORcnt.
__device__ __forceinline__ void tdm_load_tile(const float* gsrc, uint32_t lds_off,
                                              uint32_t rows) {
  const uint64_t ga = (uint64_t)(const void*)gsrc;

  // ---- D# group 0 (128b): count=1 | lds_addr | global_addr[56:0] | type=2 ----
  v4u g0;
  g0.x = 1u;                                     // count=1, user mode, no gather
  g0.y = lds_off;                                // lds_addr (bytes)
  g0.z = (uint32_t)ga;                           // global_addr[31:0]
  g0.w = ((uint32_t)(ga >> 32) & 0x01FFFFFFu)    // global_addr[56:32]
       | (2u << 30);                             // type = 2 ("image")

  // ---- D# group 1 (256b): tensor dims == tile dims (tile fully in bounds) ----
  const uint32_t dim0 = DTILE;                   // elements per row
  const uint32_t dim1 = rows;                    // rows in subtile
  const uint64_t s0   = DD;                      // row stride: 512 elements
  const uint64_t s1   = (uint64_t)TT * DD;       // unused dim1 stride (sane)
  v8i g1;
  g1.s0 = (int)(2u << 16);                                   // data_size = 4B
  g1.s1 = (int)((dim0 & 0xFFFFu) << 16);                     // abar=0 | dim0 lo16
  g1.s2 = (int)((dim0 >> 16) | ((dim1 & 0xFFFFu) << 16));    // dim0 hi | dim1 lo
  g1.s3 = (int)((dim1 >> 16) | ((uint32_t)DTILE << 16));     // dim1 hi | tile_dim0
  g1.s4 = (int)rows;                                         // tile_dim1 | tile_dim2=0
  g1.s5 = (int)(uint32_t)(s0 & 0xFFFFFFFFu);                 // dim0_stride lo32
  g1.s6 = (int)((uint32_t)(s0 >> 32) | ((uint32_t)(s1 & 0xFFFFu) << 16));
  g1.s7 = (int)(uint32_t)(s1 >> 16);                         // dim1_stride hi32

  // ---- groups 2/3: higher dims inert (tensor_dim2..4 = 1, tile dims 0) ----
  v4i g2; g2.x = 1; g2.y = 1; g2.z = 0; g2.w = 0;
  v4i g3; g3.x = 0; g3.y = (int)(1u << 16); g3.z = 0; g3.w = 0;

#if defined(__clang_major__) && (__clang_major__ >= 23)
  v8i gz = {0, 0, 0, 0, 0, 0, 0, 0};
  __builtin_amdgcn_tensor_load_to_lds(g0, g1, g2, g3, gz, 0);
#else
  __builtin_amdgcn_tensor_load_to_lds(g0, g1, g2, g3, 0);
#endif
}
#endif

// ---------------- Phase A: per-chunk local terminal value -> ws ----------------
__global__ __launch_bounds__(256) void rnn_phaseA(const float* __restrict__ x,
                                                  const float* __restrict__ w,
                                                  float* __restrict__ carry) {
  const int d = blockIdx.x * 256 + threadIdx.x;   // 0..511 (grid.x = 2)
  const int c = blockIdx.y;                       // chunk
  const int b = blockIdx.z;                       // batch
  const float a = 1.0f - fmaxf(w[d], 0.0f);
  const float* xp = x + ((size_t)b * TT + (size_t)c * LL) * DD + d;
  float h = 0.0f;
#pragma unroll 4
  for (int t = 0; t < LL; ++t) {
    __builtin_prefetch(xp + (size_t)(t + 8) * DD, 0, 0);  // global_prefetch_b8
    h = fmaf(a, h, xp[(size_t)t * DD]);
  }
  carry[((size_t)b * NC + c) * DD + d] = h;
}

// ------------- Phase B: scan chunk carries in place -> carry-in per chunk -------------
__global__ __launch_bounds__(256) void rnn_phaseB(const float* __restrict__ w,
                                                  float* __restrict__ carry) {
  const int d = blockIdx.x * 256 + threadIdx.x;   // grid.x = 2
  const int b = blockIdx.y;
  const float a = 1.0f - fmaxf(w[d], 0.0f);
  float aL = a;                                   // a^64 via 6 squarings
#pragma unroll
  for (int i = 0; i < 6; ++i) aL *= aL;
  float* p = carry + (size_t)b * NC * DD + d;
  float H = 0.0f;
#pragma unroll 4
  for (int c = 0; c < NC; ++c) {
    const float v = p[(size_t)c * DD];            // local terminal of chunk c
    p[(size_t)c * DD] = H;                        // carry-in for chunk c
    H = fmaf(aL, H, v);                           // carry-in for chunk c+1
  }
}

// ------- Phase C: double-buffered TDM tile loads, exact scan, streamed stores -------
__global__ __launch_bounds__(256) void rnn_phaseC(const float* __restrict__ x,
                                                  const float* __restrict__ w,
                                                  const float* __restrict__ hin,
                                                  float* __restrict__ out) {
  __shared__ float tile[LL * DTILE];              // 64 KB: two 32-row subtiles
  const int i  = threadIdx.x;                     // 0..255
  const int dh = blockIdx.x;                      // d half (0..1)
  const int c  = blockIdx.y;                      // chunk
  const int b  = blockIdx.z;                      // batch
  const int d  = dh * DTILE + i;

  const size_t elem0 = ((size_t)b * TT + (size_t)c * LL) * DD + (size_t)dh * DTILE;

#ifdef HAVE_TDM
  if (i < 32) {                                   // wave 0 issues both DMAs
    const uint32_t lds0 = (uint32_t)(uintptr_t)(&tile[0]);
    tdm_load_tile(x + elem0,                      lds0,                     HLL);
    tdm_load_tile(x + elem0 + (size_t)HLL * DD,   lds0 + HLL * DTILE * 4u,  HLL);
  }
  __builtin_amdgcn_s_wait_tensorcnt(1);           // subtile 0 landed (in-order)
#else
  for (int t = 0; t < HLL; ++t)
    tile[t * DTILE + i] = x[elem0 + (size_t)t * DD + i];
#endif
  __syncthreads();                                // publish subtile 0

  const float a = 1.0f - fmaxf(w[d], 0.0f);
  float h = hin[((size_t)b * NC + c) * DD + d];   // exact carry-in for this chunk
  float* op = out + elem0 + i;

#pragma unroll 4
  for (int t = 0; t < HLL; ++t) {                 // compute/store subtile 0
    h = fmaf(a, h, tile[t * DTILE + i]);          // (DMA of subtile 1 in flight)
    op[(size_t)t * DD] = h;
  }

#ifdef HAVE_TDM
  __builtin_amdgcn_s_wait_tensorcnt(0);           // subtile 1 landed
#else
  __syncthreads();
  for (int t = HLL; t < LL; ++t)
    tile[t * DTILE + i] = x[elem0 + (size_t)t * DD + i];
#endif
  __syncthreads();                                // publish subtile 1

#pragma unroll 4
  for (int t = HLL; t < LL; ++t) {                // compute/store subtile 1
    h = fmaf(a, h, tile[t * DTILE + i]);
    op[(size_t)t * DD] = h;
  }
}

extern "C" void kernel_launch(void* const* d_in, const int* in_sizes, int n_in,
                              void* d_out, int out_size, void* d_ws, size_t ws_size,
                              hipStream_t stream) {
  const float* x = (const float*)d_in[0];   // [8, 4096, 512]
  const float* w = (const float*)d_in[1];   // [512]
  float* out   = (float*)d_out;             // [8, 4096, 512]
  float* carry = (float*)d_ws;              // [8][64][512] floats = 1 MB

  dim3 blk(256);
  rnn_phaseA<<<dim3(DD / 256, NC, BB), blk, 0, stream>>>(x, w, carry);
  rnn_phaseB<<<dim3(DD / 256, BB),      blk, 0, stream>>>(w, carry);
  rnn_phaseC<<<dim3(DD / DTILE, NC, BB), blk, 0, stream>>>(x, w, carry, out);
}